// LengthRegulator_76304388980989
// MI455X (gfx1250) — compile-verified
//
#include <hip/hip_runtime.h>
#include <hip/hip_bf16.h>
#include <math.h>

typedef __attribute__((ext_vector_type(2))) float v2f;
typedef __attribute__((ext_vector_type(4))) float v4f;
typedef __attribute__((ext_vector_type(8))) float v8f;
typedef int v4i_vs __attribute__((vector_size(16)));   // matches async-LDS builtin param

#define BATCH   32
#define SEQ     1024
#define CH      384
#define KW      3
#define MAXOUT  3072
#define KDIM    (KW * CH)      // 1152
#define KC      32             // K-chunk (divides 384 -> chunk never straddles a conv tap)
#define NCHUNKS (KDIM / KC)    // 36
#define MBLK    64             // output rows per block
#define APAD    36             // A tile row stride (16B-aligned, bank-conflict free)
#define WPCHUNK ((KC / 2) * CH * 2)   // floats per repacked weight chunk (12288)
#define WPTOTAL (NCHUNKS * WPCHUNK)   // 442368 floats per layer

// CDNA5 async global->LDS staging (ASYNCcnt path), with sync fallback.
#if defined(__has_builtin)
#  if __has_builtin(__builtin_amdgcn_global_load_async_to_lds_b128)
#    define HAVE_ASYNC_LDS 1
#  endif
#endif
#ifndef HAVE_ASYNC_LDS
#  define HAVE_ASYNC_LDS 0
#endif

__device__ __forceinline__ void async_wait_all() {
#if HAVE_ASYNC_LDS
    asm volatile("s_wait_asynccnt 0x0" ::: "memory");
#endif
}

// ---------------------------------------------------------------------------
// One-time weight repack into fragment-major order:
//   wp[chunk][kpair][col][2]  with  wp[...][e] = w[chunk*32 + kpair*2 + e][col]
// so each B fragment is one aligned b64 in LDS and each chunk is a flat,
// contiguous 48 KB block (perfect for contiguous async b128 staging).
// ---------------------------------------------------------------------------
__global__ __launch_bounds__(256) void repack_w_kernel(
    const float* __restrict__ w, float* __restrict__ wp)
{
    const int i = blockIdx.x * 256 + threadIdx.x;      // pair index
    if (i >= NCHUNKS * (KC / 2) * CH) return;
    const int col = i % CH;
    const int p   = (i / CH) % (KC / 2);
    const int ch  = i / (CH * (KC / 2));
    const int k   = ch * KC + p * 2;
    wp[(size_t)i * 2 + 0] = w[(size_t)(k + 0) * CH + col];
    wp[(size_t)i * 2 + 1] = w[(size_t)(k + 1) * CH + col];
}

// ---------------------------------------------------------------------------
// Fused conv1d(K=3, SAME) + bias + LayerNorm + ReLU, optionally fused with the
// final 384->1 projection + exp (FUSE=true: writes pred[B*S] instead of h).
//
// grid.x = (B*S)/64 = 512 blocks, 256 threads (8 waves).
// Block tile: 64 rows (one batch) x all 384 channels (LayerNorm needs full H).
// Wave (mw = w&3, nw = w>>2): 16-row M tile x 12 N tiles (192 channels).
// K loop: 36 chunks of 32, double-buffered LDS staging, async B tile.
// ---------------------------------------------------------------------------
template <bool FUSE>
__global__ __launch_bounds__(256) void conv_ln_relu_kernel(
    const float* __restrict__ in,    // [B,S,384]
    const float* __restrict__ wp,    // repacked weights [36][16][384][2]
    const float* __restrict__ bias,  // [384]
    const float* __restrict__ gam,   // [384]
    const float* __restrict__ bet,   // [384]
    const float* __restrict__ wl,    // [384] (FUSE only)
    const float* __restrict__ bl,    // [1]   (FUSE only)
    float* __restrict__ out)         // !FUSE: h [B,S,384];  FUSE: pred [B,S]
{
    __shared__ float As[2][MBLK][APAD];    // 18 KB  (A, row-major, padded)
    __shared__ float Bt[2][WPCHUNK];       // 96 KB  (fragment-major B, dbl-buffered)
    __shared__ float Sred[4][16][2];       // LN sum partials   [mw][row][nw]
    __shared__ float Qred[4][16][2];       // LN sumsq partials

    const int tid  = threadIdx.x;
    const int lane = tid & 31;
    const int wave = tid >> 5;
    const int hi   = lane >> 4;            // half-wave: 0 / 1
    const int lo   = lane & 15;
    const int mw   = wave & 3;             // M sub-tile (16 rows)
    const int nw   = wave >> 2;            // N half (192 channels)

    const int m0 = blockIdx.x * MBLK;      // first global row (b*S + s)
    const int b  = m0 / SEQ;
    const int s0 = m0 % SEQ;

    v8f acc[12] = {};

    // ---- stage one K-chunk into LDS buffer nb ----
    auto stage = [&](int chunk, int nb) {
        const int k0  = chunk * KC;
        const int tap = k0 / CH;
        const int c0  = k0 % CH;
        // A: 64 rows x 32 channels (im2col w/ zero pad), 2 float4 per thread
        #pragma unroll
        for (int i = 0; i < 2; ++i) {
            const int e4  = tid + i * 256;       // 512 float4
            const int row = e4 >> 3;
            const int c4  = (e4 & 7) * 4;
            const int s   = s0 + row + tap - 1;
            v4f v = {};
            if ((unsigned)s < (unsigned)SEQ)
                v = *(const v4f*)(in + ((size_t)b * SEQ + s) * CH + c0 + c4);
            *(v4f*)&As[nb][row][c4] = v;
        }
        // B: one flat 48 KB fragment-major chunk, 12 contiguous float4/thread
        const float* srcBase = wp + (size_t)chunk * WPCHUNK;
        #pragma unroll
        for (int i = 0; i < 12; ++i) {
            const int f4 = tid + i * 256;        // 3072 float4
#if HAVE_ASYNC_LDS
            __builtin_amdgcn_global_load_async_to_lds_b128(
                (v4i_vs*)(srcBase + f4 * 4), (v4i_vs*)&Bt[nb][f4 * 4], 0, 0);
#else
            *(v4f*)&Bt[nb][f4 * 4] = *(const v4f*)(srcBase + f4 * 4);
#endif
        }
    };

    // ---- 8 k-steps x 12 n-tiles of v_wmma_f32_16x16x4_f32 per wave ----
    auto compute_chunk = [&](int cb) {
        const int colbase = nw * 192 + lo;
        #pragma unroll
        for (int kk = 0; kk < KC; kk += 4) {
            // A frag (ISA 7.12.2): VGPR0 = K{0|2}, VGPR1 = K{1|3} per lane half
            v2f a = *(const v2f*)&As[cb][mw * 16 + lo][kk + 2 * hi];
            const float* bp = &Bt[cb][((kk / 2 + hi) * CH + colbase) * 2];
            #pragma unroll
            for (int t = 0; t < 12; ++t) {
                v2f bb = *(const v2f*)(bp + t * 32);   // (col + 16)*2 per tile
                acc[t] = __builtin_amdgcn_wmma_f32_16x16x4_f32(
                    false, a, false, bb, (short)0, acc[t], false, false);
            }
        }
    };

    int buf = 0;
    stage(0, 0);
    async_wait_all();
    __syncthreads();
    for (int chunk = 0; chunk < NCHUNKS; ++chunk) {
        if (chunk + 1 < NCHUNKS) stage(chunk + 1, buf ^ 1);
        compute_chunk(buf);
        async_wait_all();      // drain next buffer's async loads
        __syncthreads();
        buf ^= 1;
    }

    // ================= epilogue: bias + LayerNorm (+ ReLU) =================
    const int rowbase = m0 + mw * 16;

    float psum[8] = {}, pssq[8] = {};
    #pragma unroll
    for (int t = 0; t < 12; ++t) {
        const int col = nw * 192 + t * 16 + lo;
        const float bv = bias[col];
        #pragma unroll
        for (int j = 0; j < 8; ++j) {
            const float v = acc[t][j] + bv;
            acc[t][j] = v;
            psum[j] += v;
            pssq[j] += v * v;
        }
    }
    // reduce across the 16 lanes of each half-wave (they share rows j+8*hi)
    #pragma unroll
    for (int m = 1; m <= 8; m <<= 1) {
        #pragma unroll
        for (int j = 0; j < 8; ++j) {
            psum[j] += __shfl_xor(psum[j], m, 32);
            pssq[j] += __shfl_xor(pssq[j], m, 32);
        }
    }
    if (lo == 0) {
        #pragma unroll
        for (int j = 0; j < 8; ++j) {
            Sred[mw][j + 8 * hi][nw] = psum[j];
            Qred[mw][j + 8 * hi][nw] = pssq[j];
        }
    }
    __syncthreads();
    float mu[8], inv[8];
    #pragma unroll
    for (int j = 0; j < 8; ++j) {
        const int r = j + 8 * hi;
        const float s = Sred[mw][r][0] + Sred[mw][r][1];
        const float q = Qred[mw][r][0] + Qred[mw][r][1];
        const float m_ = s * (1.0f / CH);
        mu[j]  = m_;
        inv[j] = rsqrtf(q * (1.0f / CH) - m_ * m_ + 1e-5f);
    }
    __syncthreads();   // stats consumed; Sred reusable below

    if (!FUSE) {
        #pragma unroll
        for (int t = 0; t < 12; ++t) {
            const int col = nw * 192 + t * 16 + lo;
            const float gv = gam[col], bv = bet[col];
            #pragma unroll
            for (int j = 0; j < 8; ++j) {
                float v = (acc[t][j] - mu[j]) * inv[j] * gv + bv;
                v = v > 0.f ? v : 0.f;
                out[((size_t)rowbase + j + 8 * hi) * CH + col] = v;
            }
        }
    } else {
        // fused 384->1 projection + exp: h2 never touches memory
        float pdot[8] = {};
        #pragma unroll
        for (int t = 0; t < 12; ++t) {
            const int col = nw * 192 + t * 16 + lo;
            const float gv = gam[col], bv = bet[col], wv = wl[col];
            #pragma unroll
            for (int j = 0; j < 8; ++j) {
                float v = (acc[t][j] - mu[j]) * inv[j] * gv + bv;
                v = v > 0.f ? v : 0.f;
                pdot[j] += v * wv;
            }
        }
        #pragma unroll
        for (int m = 1; m <= 8; m <<= 1) {
            #pragma unroll
            for (int j = 0; j < 8; ++j)
                pdot[j] += __shfl_xor(pdot[j], m, 32);
        }
        if (lo == 0) {
            #pragma unroll
            for (int j = 0; j < 8; ++j)
                Sred[mw][j + 8 * hi][nw] = pdot[j];
        }
        __syncthreads();
        if (nw == 0 && lo == 0) {
            const float bl0 = bl[0];
            #pragma unroll
            for (int j = 0; j < 8; ++j) {
                const int r = j + 8 * hi;
                out[rowbase + r] = expf(Sred[mw][r][0] + Sred[mw][r][1] + bl0);
            }
        }
    }
}

// ---------------------------------------------------------------------------
// Inclusive cumsum of durations, one block (256 threads, 4 elems each) per batch.
// ---------------------------------------------------------------------------
__global__ __launch_bounds__(256) void cumsum_kernel(
    const int* __restrict__ dur, int* __restrict__ cum)
{
    __shared__ int part[256];
    const int b   = blockIdx.x;
    const int tid = threadIdx.x;
    const int* d = dur + (size_t)b * SEQ;
    int*       c = cum + (size_t)b * SEQ;

    int loc[4];
    int s = 0;
    #pragma unroll
    for (int i = 0; i < 4; ++i) { s += d[tid * 4 + i]; loc[i] = s; }
    part[tid] = s;
    __syncthreads();
    for (int off = 1; off < 256; off <<= 1) {
        int v = (tid >= off) ? part[tid - off] : 0;
        __syncthreads();
        part[tid] += v;
        __syncthreads();
    }
    const int base = (tid > 0) ? part[tid - 1] : 0;
    #pragma unroll
    for (int i = 0; i < 4; ++i) c[tid * 4 + i] = base + loc[i];
}

// ---------------------------------------------------------------------------
// Length regulation: one wave per output row (b,t). searchsorted-right on cum,
// gather x row, mask invalid tail. Streams the 151 MB output.
// ---------------------------------------------------------------------------
__global__ __launch_bounds__(256) void regulate_kernel(
    const float* __restrict__ x, const int* __restrict__ cum,
    float* __restrict__ res)
{
    const int idx  = blockIdx.x * 8 + (threadIdx.x >> 5);  // [0, B*MAXOUT)
    const int lane = threadIdx.x & 31;
    const int b = idx / MAXOUT;
    const int t = idx % MAXOUT;
    const int* c = cum + (size_t)b * SEQ;

    const float valid = (t < c[SEQ - 1]) ? 1.f : 0.f;
    int loi = 0, hii = SEQ;
    while (loi < hii) {
        const int mid = (loi + hii) >> 1;
        if (c[mid] <= t) loi = mid + 1; else hii = mid;
    }
    const int src = loi < (SEQ - 1) ? loi : (SEQ - 1);

    const float* xr = x + ((size_t)b * SEQ + src) * CH;
    float* orow = res + (size_t)idx * CH;
    #pragma unroll
    for (int j = 0; j < 3; ++j) {
        v4f v = *(const v4f*)(xr + lane * 4 + j * 128);
        *(v4f*)(orow + lane * 4 + j * 128) = v * valid;
    }
}

// ---------------------------------------------------------------------------
extern "C" void kernel_launch(void* const* d_in, const int* in_sizes, int n_in,
                              void* d_out, int out_size, void* d_ws, size_t ws_size,
                              hipStream_t stream) {
    (void)in_sizes; (void)n_in; (void)out_size; (void)ws_size;
    const float* x   = (const float*)d_in[0];
    const int*   dur = (const int*)  d_in[1];
    const float* w1  = (const float*)d_in[2];
    const float* b1  = (const float*)d_in[3];
    const float* g1  = (const float*)d_in[4];
    const float* be1 = (const float*)d_in[5];
    const float* w2  = (const float*)d_in[6];
    const float* b2  = (const float*)d_in[7];
    const float* g2  = (const float*)d_in[8];
    const float* be2 = (const float*)d_in[9];
    const float* wl  = (const float*)d_in[10];
    const float* bl  = (const float*)d_in[11];

    float* res  = (float*)d_out;                                   // [B, MAXOUT, CH]
    float* pred = res + (size_t)BATCH * MAXOUT * CH;               // [B, S]

    // h1 scratch lives inside the res region: regulate_kernel rewrites every
    // element of res last, so this is safe and deterministic across replays.
    float* h1 = res;                                               // B*S*CH floats

    // workspace: repacked weights (2 x 1.77 MB) + cumsum (128 KB)
    float* wp1 = (float*)d_ws;
    float* wp2 = wp1 + WPTOTAL;
    int*   cum = (int*)(wp2 + WPTOTAL);

    const int nPairs   = NCHUNKS * (KC / 2) * CH;
    const int repackGr = (nPairs + 255) / 256;

    repack_w_kernel<<<repackGr, 256, 0, stream>>>(w1, wp1);
    repack_w_kernel<<<repackGr, 256, 0, stream>>>(w2, wp2);
    conv_ln_relu_kernel<false><<<(BATCH * SEQ) / MBLK, 256, 0, stream>>>(
        x, wp1, b1, g1, be1, nullptr, nullptr, h1);
    conv_ln_relu_kernel<true><<<(BATCH * SEQ) / MBLK, 256, 0, stream>>>(
        h1, wp2, b2, g2, be2, wl, bl, pred);
    cumsum_kernel  <<<BATCH,                256, 0, stream>>>(dur, cum);
    regulate_kernel<<<(BATCH * MAXOUT) / 8, 256, 0, stream>>>(x, cum, res);
}